// connect_attention_59090160058553
// MI455X (gfx1250) — compile-verified
//
#include <hip/hip_runtime.h>
#include <math.h>
#include <stdint.h>

typedef __attribute__((ext_vector_type(2))) float v2f;
typedef __attribute__((ext_vector_type(8))) float v8f;

#define NN     16384
#define LIMIT  8192
#define KW     7

// Block-wide inclusive scan over 1024 values (ping-pong in buf[0..2047]).
__device__ __forceinline__ unsigned block_scan_inclusive(unsigned val,
                                                         unsigned* buf,
                                                         int tid) {
  buf[tid] = val;
  __syncthreads();
  int src = 0;
  #pragma unroll
  for (int off = 1; off < 1024; off <<= 1) {
    unsigned v = buf[src * 1024 + tid];
    if (tid >= off) v += buf[src * 1024 + tid - off];
    buf[(src ^ 1) * 1024 + tid] = v;
    src ^= 1;
    __syncthreads();
  }
  return buf[src * 1024 + tid];
}

// Single-workgroup fused kernel: async-copy x into LDS, conv7 via chained
// V_WMMA_F32_16X16X4_F32, sigmoid, parallel radix-select of the bottom half,
// stable tie-break, masked scale + scatter. 1024 threads = 32 wave32s on one WGP.
__launch_bounds__(1024)
__global__ void fused_connect_attention(const float* __restrict__ x,
                                        const float* __restrict__ w,
                                        float* __restrict__ out) {
  __shared__ __align__(16) float xs[NN + 32];  // xs[4+i] = x[i], zero halos
  __shared__ unsigned sbits[NN];               // sigmoid scores as monotone u32
  __shared__ unsigned hist[2048];              // radix histogram
  __shared__ unsigned scanbuf[2048];           // scan ping-pong
  __shared__ float    wt[24 * 16];             // Toeplitz band W[j][n]
  __shared__ unsigned sh_prefix;
  __shared__ int      sh_r;

  const int tid  = (int)threadIdx.x;
  const int lane = tid & 31;
  const int wid  = tid >> 5;

  // ---- async-stage x into padded LDS: 4 x b128 per thread (CDNA5 async path) ----
  {
    const unsigned xs_lds = (unsigned)(size_t)(&xs[0]);
    const uint64_t xbase  = (uint64_t)(size_t)x;
    #pragma unroll
    for (int e = 0; e < 4; ++e) {
      int      c    = tid + 1024 * e;                 // float4 chunk id
      unsigned voff = (unsigned)(c * 16);             // global byte offset
      unsigned ldst = xs_lds + 16u + (unsigned)(c * 16);  // LDS byte addr (xs[4+4c])
      asm volatile("global_load_async_to_lds_b128 %0, %1, %2"
                   :: "v"(ldst), "v"(voff), "s"(xbase) : "memory");
    }
  }

  if (tid < 4)  xs[tid] = 0.0f;
  if (tid < 28) xs[4 + NN + tid] = 0.0f;
  if (tid == 0) { sh_prefix = 0u; sh_r = LIMIT - 1; }

  // ---- Toeplitz band in LDS: wt[j*16+n] = w[j-n] if 0 <= j-n < 7 else 0 ----
  if (tid < 24 * 16) {
    int j = tid >> 4, nn = tid & 15;
    int d = j - nn;
    wt[tid] = (d >= 0 && d < KW) ? w[d] : 0.0f;
  }

  asm volatile("s_wait_asynccnt 0" ::: "memory");
  __syncthreads();

  // ---- per-lane WMMA operand mapping (ISA 32-bit layouts) ----
  // lanes 0-15 hold K={0,1} in VGPRs {0,1}; lanes 16-31 hold K={2,3}.
  const int n    = lane & 15;   // B col / A row
  const int half = lane >> 4;
  v2f bv[6];
  #pragma unroll
  for (int kc = 0; kc < 6; ++kc) {
    int j0 = 4 * kc + 2 * half;
    bv[kc].x = wt[j0 * 16 + n];
    bv[kc].y = wt[(j0 + 1) * 16 + n];
  }

  // ---- conv7 + sigmoid: y[base+16m+n] = sum_j A[m][j] * W[j][n] ----
  #pragma unroll
  for (int t = 0; t < 2; ++t) {
    const int base = (wid * 2 + t) * 256;
    v8f acc = {};
    #pragma unroll
    for (int kc = 0; kc < 6; ++kc) {
      int j0 = 4 * kc + 2 * half;
      int xi = base + 16 * n + j0 - 3;   // A[m][j] = x[base + 16m + j - 3]
      v2f a;
      a.x = xs[4 + xi];
      a.y = xs[4 + xi + 1];
      acc = __builtin_amdgcn_wmma_f32_16x16x4_f32(
          false, a, false, bv[kc], (short)0, acc, false, false);
    }
    // C/D layout: VGPR r -> M = r + 8*half, N = lane & 15
    #pragma unroll
    for (int r = 0; r < 8; ++r) {
      int i   = base + 16 * r + 128 * half + n;
      float s = 1.0f / (1.0f + expf(-acc[r]));
      sbits[i] = __float_as_uint(s);     // positive floats: bits are monotone
    }
  }
  __syncthreads();

  // ---- radix-select rank (LIMIT-1): 11/11/10-bit passes, parallel bin pick ----
  const int shifts[3] = {21, 10, 0};
  const int nbits[3]  = {11, 11, 10};
  unsigned fixedmask = 0u;
  #pragma unroll
  for (int p = 0; p < 3; ++p) {
    const int      shift = shifts[p];
    const int      nbins = 1 << nbits[p];
    const unsigned dmask = (unsigned)(nbins - 1);
    for (int b = tid; b < nbins; b += 1024) hist[b] = 0u;
    __syncthreads();
    const unsigned pfx = sh_prefix;
    const unsigned r   = (unsigned)sh_r;
    #pragma unroll
    for (int e = 0; e < 16; ++e) {
      unsigned b = sbits[tid + 1024 * e];
      if ((b & fixedmask) == pfx)
        atomicAdd(&hist[(b >> shift) & dmask], 1u);
    }
    __syncthreads();
    // fold this thread's bins, block-scan, then test own bins for the rank
    const int bpt = nbins >> 10;             // bins per thread: 2,2,1
    unsigned c = 0u;
    for (int j = 0; j < bpt; ++j) c += hist[tid * bpt + j];
    unsigned incl    = block_scan_inclusive(c, scanbuf, tid);
    unsigned running = incl - c;             // exclusive prefix over bins
    for (int j = 0; j < bpt; ++j) {
      unsigned cnt = hist[tid * bpt + j];
      if (r >= running && r < running + cnt) {
        sh_prefix = pfx | ((unsigned)(tid * bpt + j) << shift);
        sh_r      = (int)(r - running);
      }
      running += cnt;
    }
    fixedmask |= dmask << shift;
    __syncthreads();
  }

  const unsigned T    = sh_prefix;   // rank-(LIMIT-1) score bits (last kept)
  const int      need = sh_r + 1;    // elements equal to T to keep (by index)

  // ---- stable tie-break: index-ordered exclusive prefix of (bits == T) ----
  const int cbase = tid * 16;
  unsigned cnt = 0u;
  #pragma unroll
  for (int e = 0; e < 16; ++e) cnt += (sbits[cbase + e] == T) ? 1u : 0u;
  unsigned incl = block_scan_inclusive(cnt, scanbuf, tid);
  int rank = (int)(incl - cnt);

  // ---- emit new_x and score as float4 stores ----
  #pragma unroll
  for (int q = 0; q < 4; ++q) {
    float4 nx, sc;
    #pragma unroll
    for (int e2 = 0; e2 < 4; ++e2) {
      int      i = cbase + q * 4 + e2;
      unsigned b = sbits[i];
      float    s = __uint_as_float(b);
      bool keep;
      if (b == T) { keep = (rank < need); ++rank; }
      else        { keep = (b < T); }
      (&nx.x)[e2] = keep ? xs[4 + i] * (s + 1.0f) : 0.0f;
      (&sc.x)[e2] = s;
    }
    reinterpret_cast<float4*>(out + cbase)[q]      = nx;
    reinterpret_cast<float4*>(out + NN + cbase)[q] = sc;
  }
}

extern "C" void kernel_launch(void* const* d_in, const int* in_sizes, int n_in,
                              void* d_out, int out_size, void* d_ws, size_t ws_size,
                              hipStream_t stream) {
  const float* x = (const float*)d_in[0];
  const float* w = (const float*)d_in[1];
  float* out = (float*)d_out;
  (void)in_sizes; (void)n_in; (void)out_size; (void)d_ws; (void)ws_size;
  hipLaunchKernelGGL(fused_connect_attention, dim3(1), dim3(1024), 0, stream,
                     x, w, out);
}